// BayesTensorRing_81303730913631
// MI455X (gfx1250) — compile-verified
//
#include <hip/hip_runtime.h>

typedef __attribute__((ext_vector_type(2))) float v2f;
typedef __attribute__((ext_vector_type(8))) float v8f;
typedef __attribute__((ext_vector_type(4))) int   v4i;

// One wave32 per sample (grid-stride). Computes trace(M0' M1' M2' M3') with
// M_d' = core_d[idx_d] * diag(lam_d), via two independent fp32 WMMA chains:
//   P  = M0 * (L0 M1 L1)            (16x16x16 = 4x V_WMMA_F32_16X16X4_F32)
//   Q' = (L3 M3^T) * (L2 M2^T)      (= Q^T, same C/D layout as P)
// out = sum_{m,n} P[m,n] * Q[n,m] = elementwise dot of P and Q' registers.
//
// Indices are wave-uniform; readfirstlane forces them into SGPRs so all 24
// fragment gathers use saddr + loop-invariant 32-bit VGPR offsets.
__global__ __launch_bounds__(256) void tr_trace_wmma_kernel(
    const long long* __restrict__ index,
    const float* __restrict__ core0, const float* __restrict__ lam0,
    const float* __restrict__ core1, const float* __restrict__ lam1,
    const float* __restrict__ core2, const float* __restrict__ lam2,
    const float* __restrict__ core3, const float* __restrict__ lam3,
    float* __restrict__ out, int n)
{
    const int lane = threadIdx.x & 31;
    const int waves_per_block = blockDim.x >> 5;
    const int wave   = blockIdx.x * waves_per_block + (threadIdx.x >> 5);
    const int nwaves = gridDim.x * waves_per_block;

    const int nm = lane & 15;   // row/col index m or n (0..15) held by this lane
    const int h  = lane >> 4;   // half selector: K-subtile within a WMMA chunk

    // ---- hoisted, sample-independent lambda factors ----
    const float l1n = lam1[nm];  // column scale of P's B operand (lane-uniform)
    const float l3m = lam3[nm];  // row scale of Q's A operand (lane-uniform)
    v2f l0p[4], l2p[4];          // per-chunk K-pair scales
#pragma unroll
    for (int c = 0; c < 4; ++c) {
        const int k = 4 * c + 2 * h;
        l0p[c].x = lam0[k];  l0p[c].y = lam0[k + 1];
        l2p[c].x = lam2[k];  l2p[c].y = lam2[k + 1];
    }

    // ---- loop-invariant per-lane element offsets (in floats) ----
    int offA[4], offBx[4], offBy[4];
#pragma unroll
    for (int c = 0; c < 4; ++c) {
        const int kk = 4 * c + 2 * h;
        offA[c]  = nm * 16 + kk;        // row-pair load (b64)
        offBx[c] = (kk    ) * 16 + nm;  // column-walk loads (b32)
        offBy[c] = (kk + 1) * 16 + nm;
    }

    for (int s = wave; s < n; s += nwaves) {
        // s is wave-uniform by construction; make it provably so.
        const int su = __builtin_amdgcn_readfirstlane(s);

        // index row = 32B aligned; fetch as two b128, keep low dwords
        // (values < 200 fit in 32 bits), pin to SGPRs.
        const v4i q0 = *(const v4i*)(index + 4 * (long long)su);
        const v4i q1 = *(const v4i*)(index + 4 * (long long)su + 2);
        const int i0 = __builtin_amdgcn_readfirstlane(q0.x);
        const int i1 = __builtin_amdgcn_readfirstlane(q0.z);
        const int i2 = __builtin_amdgcn_readfirstlane(q1.x);
        const int i3 = __builtin_amdgcn_readfirstlane(q1.z);
        const float* M0 = core0 + (long long)i0 * 256;
        const float* M1 = core1 + (long long)i1 * 256;
        const float* M2 = core2 + (long long)i2 * 256;
        const float* M3 = core3 + (long long)i3 * 256;

        if (su + nwaves < n)
            __builtin_prefetch(index + 4 * (long long)(su + nwaves), 0, 0);

        // ---- issue all fragment loads up front (latency hiding) ----
        v2f a0[4], b0[4], a1[4], b1[4];
#pragma unroll
        for (int c = 0; c < 4; ++c) {
            // P chain: A[m][k] = M0[m][k]           (aligned b64 row load)
            a0[c] = *(const v2f*)(M0 + offA[c]);
            // P chain: B[k][n] = lam0[k]*M1[k][n]*lam1[n]
            b0[c].x = M1[offBx[c]] * (l0p[c].x * l1n);
            b0[c].y = M1[offBy[c]] * (l0p[c].y * l1n);
            // Q' chain: A[m][k] = lam3[m]*M3[k][m]
            a1[c].x = M3[offBx[c]] * l3m;
            a1[c].y = M3[offBy[c]] * l3m;
            // Q' chain: B[k][n] = lam2[k]*M2[n][k]  (aligned b64 row load)
            v2f r = *(const v2f*)(M2 + offA[c]);
            b1[c].x = r.x * l2p[c].x;
            b1[c].y = r.y * l2p[c].y;
        }

        // ---- two independent 4-deep WMMA accumulation chains ----
        v8f P = {0.f, 0.f, 0.f, 0.f, 0.f, 0.f, 0.f, 0.f};
        v8f Q = {0.f, 0.f, 0.f, 0.f, 0.f, 0.f, 0.f, 0.f};
#pragma unroll
        for (int c = 0; c < 4; ++c) {
            P = __builtin_amdgcn_wmma_f32_16x16x4_f32(
                    false, a0[c], false, b0[c], (short)0, P, false, false);
            Q = __builtin_amdgcn_wmma_f32_16x16x4_f32(
                    false, a1[c], false, b1[c], (short)0, Q, false, false);
        }

        // ---- trace: per-lane dot over the 8 accumulator VGPRs ----
        float t = 0.f;
#pragma unroll
        for (int v = 0; v < 8; ++v) t = fmaf(P[v], Q[v], t);

        // wave32 butterfly reduction (log2(32) = 5 steps)
#pragma unroll
        for (int off = 16; off > 0; off >>= 1)
            t += __shfl_xor(t, off, 32);

        if (lane == 0) out[su] = t;
    }
}

extern "C" void kernel_launch(void* const* d_in, const int* in_sizes, int n_in,
                              void* d_out, int out_size, void* d_ws, size_t ws_size,
                              hipStream_t stream) {
    // setup_inputs() order: index, core0, lam0, core1, lam1, core2, lam2, core3, lam3
    const long long* index = (const long long*)d_in[0];
    const float* core0 = (const float*)d_in[1];
    const float* lam0  = (const float*)d_in[2];
    const float* core1 = (const float*)d_in[3];
    const float* lam1  = (const float*)d_in[4];
    const float* core2 = (const float*)d_in[5];
    const float* lam2  = (const float*)d_in[6];
    const float* core3 = (const float*)d_in[7];
    const float* lam3  = (const float*)d_in[8];
    float* out = (float*)d_out;

    const int n = in_sizes[0] / 4;   // N samples (index is N x 4 int64)

    // 4096 blocks x 8 wave32 = 32768 waves; ~15 samples per wave amortizes
    // the hoisted lambda/offset setup and keeps the machine saturated.
    dim3 grid(4096), block(256);
    tr_trace_wmma_kernel<<<grid, block, 0, stream>>>(
        index, core0, lam0, core1, lam1, core2, lam2, core3, lam3, out, n);
}